// REINFORCE_graph_72241349919439
// MI455X (gfx1250) — compile-verified
//
#include <hip/hip_runtime.h>
#include <hip/hip_bf16.h>

#define N_NODES   50000
#define N_EDGES   800000
#define HIDDEN    128
#define IN_DIM    6
#define N_ACTIONS 64
#define NUM_GRAPHS 512

typedef __attribute__((ext_vector_type(2))) float v2f;
typedef __attribute__((ext_vector_type(8))) float v8f;

// ---------------------------------------------------------------------------
// 1) init: deg = 1 (self loop), gsum = 0, cnt = 0
// ---------------------------------------------------------------------------
__global__ void k_init(float* deg, float* gsum, float* cnt) {
    int i = blockIdx.x * blockDim.x + threadIdx.x;
    if (i < N_NODES)            deg[i]  = 1.0f;
    if (i < NUM_GRAPHS*HIDDEN)  gsum[i] = 0.0f;
    if (i < NUM_GRAPHS)         cnt[i]  = 0.0f;
}

// ---------------------------------------------------------------------------
// 2) degree accumulation over dst (edge_index row 1)
// ---------------------------------------------------------------------------
__global__ void k_deg(const int* __restrict__ ei, float* deg) {
    int e = blockIdx.x * blockDim.x + threadIdx.x;
    if (e < N_EDGES) atomicAdd(&deg[ei[N_EDGES + e]], 1.0f);
}

// ---------------------------------------------------------------------------
// 3) dinv = rsqrt(max(deg,1))   (deg >= 1 from self-loop)
// ---------------------------------------------------------------------------
__global__ void k_dinv(float* deg) {
    int i = blockIdx.x * blockDim.x + threadIdx.x;
    if (i < N_NODES) deg[i] = rsqrtf(fmaxf(deg[i], 1.0f));
}

// ---------------------------------------------------------------------------
// 4) h_lin = x @ w1 ;  h_agg seeded with self-loop term + b1
//    one block (128 threads) per node, thread = output channel
// ---------------------------------------------------------------------------
__global__ void k_lin1(const float* __restrict__ x, const float* __restrict__ w1,
                       const float* __restrict__ b1, const float* __restrict__ dinv,
                       float* __restrict__ h_lin, float* __restrict__ h_agg) {
    int n = blockIdx.x;
    int c = threadIdx.x;
    __shared__ float xs[IN_DIM];
    if (c < IN_DIM) xs[c] = x[n * IN_DIM + c];
    __syncthreads();
    float s = 0.0f;
#pragma unroll
    for (int k = 0; k < IN_DIM; ++k) s += xs[k] * w1[k * HIDDEN + c];
    h_lin[n * HIDDEN + c] = s;
    float di = dinv[n];
    h_agg[n * HIDDEN + c] = s * di * di + b1[c];   // self-loop norm = dinv^2
}

// ---------------------------------------------------------------------------
// 5) edge scatter: one wave32 per edge, float4 per lane (128 ch)
//    h_agg[dst] += h_lin[src] * dinv[src]*dinv[dst]
// ---------------------------------------------------------------------------
__global__ void k_edge(const int* __restrict__ ei, const float* __restrict__ dinv,
                       const float* __restrict__ h_lin, float* __restrict__ h_agg) {
    int gid  = blockIdx.x * blockDim.x + threadIdx.x;
    int e    = gid >> 5;
    int lane = gid & 31;
    if (e >= N_EDGES) return;
    int s = ei[e];
    int d = ei[N_EDGES + e];
    float norm = dinv[s] * dinv[d];
    const float4* hs = (const float4*)(h_lin + (size_t)s * HIDDEN);
    float4 v = hs[lane];
    float* o = h_agg + (size_t)d * HIDDEN + lane * 4;
    atomicAdd(o + 0, v.x * norm);
    atomicAdd(o + 1, v.y * norm);
    atomicAdd(o + 2, v.z * norm);
    atomicAdd(o + 3, v.w * norm);
}

// ---------------------------------------------------------------------------
// 6) pooling: gsum[batch[n]] += relu(h_agg[n]); cnt[batch[n]] += 1
//    one wave32 per node, float4 per lane
// ---------------------------------------------------------------------------
__global__ void k_pool(const int* __restrict__ batch, const float* __restrict__ h_agg,
                       float* __restrict__ gsum, float* __restrict__ cnt) {
    int gid  = blockIdx.x * blockDim.x + threadIdx.x;
    int n    = gid >> 5;
    int lane = gid & 31;
    if (n >= N_NODES) return;
    int g = batch[n];
    const float4* h = (const float4*)(h_agg + (size_t)n * HIDDEN);
    float4 v = h[lane];
    float* o = gsum + (size_t)g * HIDDEN + lane * 4;
    atomicAdd(o + 0, fmaxf(v.x, 0.0f));
    atomicAdd(o + 1, fmaxf(v.y, 0.0f));
    atomicAdd(o + 2, fmaxf(v.z, 0.0f));
    atomicAdd(o + 3, fmaxf(v.w, 0.0f));
    if (lane == 0) atomicAdd(&cnt[g], 1.0f);
}

// ---------------------------------------------------------------------------
// 7) mean: g = gsum / max(cnt, 1)
// ---------------------------------------------------------------------------
__global__ void k_mean(const float* __restrict__ gsum, const float* __restrict__ cnt,
                       float* __restrict__ g) {
    int i = blockIdx.x * blockDim.x + threadIdx.x;
    if (i < NUM_GRAPHS * HIDDEN) {
        int r = i / HIDDEN;
        g[i] = gsum[i] / fmaxf(cnt[r], 1.0f);
    }
}

// ---------------------------------------------------------------------------
// WMMA f32 16x16x4 fragment layout (ISA 7.12.2):
//   A (16x4): lanes 0-15 -> M=lane, VGPR0=K0,VGPR1=K1; lanes 16-31 -> K2,K3
//   B (4x16): lanes 0-15 -> N=lane, VGPR0=K0,VGPR1=K1; lanes 16-31 -> K2,K3
//   C/D: lane<16 -> N=lane, VGPR v = row M=v; lane>=16 -> N=lane-16, M=v+8
// ---------------------------------------------------------------------------

// 8) g2 = relu(g @ wl + bl)   M=512 K=128 N=128, one 16x16 tile per wave
__global__ void k_mlp1(const float* __restrict__ g, const float* __restrict__ wl,
                       const float* __restrict__ bl, float* __restrict__ g2) {
    int t    = blockIdx.x * (blockDim.x >> 5) + (threadIdx.x >> 5);
    int mt   = t >> 3;           // 0..31
    int nt   = t & 7;            // 0..7
    int m0   = mt * 16, n0 = nt * 16;
    int lane = threadIdx.x & 31;
    int lr   = lane & 15;
    int hi   = lane >> 4;        // 0: K pair {0,1}, 1: K pair {2,3}
    v8f acc  = {};
    for (int kk = 0; kk < HIDDEN / 4; ++kk) {
        int k0 = kk * 4 + 2 * hi;
        v2f a, b;
        a.x = g[(m0 + lr) * HIDDEN + k0];
        a.y = g[(m0 + lr) * HIDDEN + k0 + 1];
        b.x = wl[(k0    ) * HIDDEN + n0 + lr];
        b.y = wl[(k0 + 1) * HIDDEN + n0 + lr];
        acc = __builtin_amdgcn_wmma_f32_16x16x4_f32(false, a, false, b,
                                                    (short)0, acc, false, false);
    }
    int   col  = n0 + lr;
    float bias = bl[col];
#pragma unroll
    for (int v = 0; v < 8; ++v) {
        int row = m0 + hi * 8 + v;
        g2[row * HIDDEN + col] = fmaxf(acc[v] + bias, 0.0f);
    }
}

// 9) logits = g2 @ w2 + b2, then row-wise log_softmax -> out
//    one block = 16 rows; 4 waves = 4 column tiles of 16 (covers 64 actions)
__global__ void k_head(const float* __restrict__ g2, const float* __restrict__ w2,
                       const float* __restrict__ b2, float* __restrict__ out) {
    __shared__ float lg[16][N_ACTIONS];
    int wave = threadIdx.x >> 5;
    int lane = threadIdx.x & 31;
    int m0   = blockIdx.x * 16;
    int n0   = wave * 16;
    int lr   = lane & 15;
    int hi   = lane >> 4;
    v8f acc  = {};
    for (int kk = 0; kk < HIDDEN / 4; ++kk) {
        int k0 = kk * 4 + 2 * hi;
        v2f a, b;
        a.x = g2[(m0 + lr) * HIDDEN + k0];
        a.y = g2[(m0 + lr) * HIDDEN + k0 + 1];
        b.x = w2[(k0    ) * N_ACTIONS + n0 + lr];
        b.y = w2[(k0 + 1) * N_ACTIONS + n0 + lr];
        acc = __builtin_amdgcn_wmma_f32_16x16x4_f32(false, a, false, b,
                                                    (short)0, acc, false, false);
    }
    int   col  = n0 + lr;
    float bias = b2[col];
#pragma unroll
    for (int v = 0; v < 8; ++v)
        lg[hi * 8 + v][col] = acc[v] + bias;
    __syncthreads();

    // log_softmax: one thread per row (16 rows, 64 cols each)
    if (threadIdx.x < 16) {
        int r = threadIdx.x;
        float m = lg[r][0];
        for (int c = 1; c < N_ACTIONS; ++c) m = fmaxf(m, lg[r][c]);
        float s = 0.0f;
        for (int c = 0; c < N_ACTIONS; ++c) s += __expf(lg[r][c] - m);
        float lse = m + __logf(s);
        for (int c = 0; c < N_ACTIONS; ++c)
            out[(m0 + r) * N_ACTIONS + c] = lg[r][c] - lse;
    }
}

// ---------------------------------------------------------------------------
extern "C" void kernel_launch(void* const* d_in, const int* in_sizes, int n_in,
                              void* d_out, int out_size, void* d_ws, size_t ws_size,
                              hipStream_t stream) {
    const float* x     = (const float*)d_in[0];
    const int*   ei    = (const int*)  d_in[1];   // [2, E] flat: row0 = src, row1 = dst
    const int*   batch = (const int*)  d_in[2];
    const float* w1    = (const float*)d_in[3];
    const float* b1    = (const float*)d_in[4];
    const float* wl    = (const float*)d_in[5];
    const float* bl    = (const float*)d_in[6];
    const float* w2    = (const float*)d_in[7];
    const float* b2    = (const float*)d_in[8];
    float* out = (float*)d_out;

    // workspace layout (256B aligned chunks)
    char* ws = (char*)d_ws;
    size_t off = 0;
    auto* deg   = (float*)(ws + off); off += ((size_t)N_NODES * 4 + 255) & ~(size_t)255;
    auto* h_lin = (float*)(ws + off); off += (size_t)N_NODES * HIDDEN * 4;
    auto* h_agg = (float*)(ws + off); off += (size_t)N_NODES * HIDDEN * 4;
    auto* gsum  = (float*)(ws + off); off += (size_t)NUM_GRAPHS * HIDDEN * 4;
    auto* cnt   = (float*)(ws + off); off += ((size_t)NUM_GRAPHS * 4 + 255) & ~(size_t)255;
    auto* gmean = (float*)(ws + off); off += (size_t)NUM_GRAPHS * HIDDEN * 4;
    auto* g2    = (float*)(ws + off); off += (size_t)NUM_GRAPHS * HIDDEN * 4;

    int initN = NUM_GRAPHS * HIDDEN;               // 65536 > N_NODES
    k_init<<<(initN + 255) / 256, 256, 0, stream>>>(deg, gsum, cnt);
    k_deg <<<(N_EDGES + 255) / 256, 256, 0, stream>>>(ei, deg);
    k_dinv<<<(N_NODES + 255) / 256, 256, 0, stream>>>(deg);
    k_lin1<<<N_NODES, HIDDEN, 0, stream>>>(x, w1, b1, deg, h_lin, h_agg);
    k_edge<<<(N_EDGES * 32 + 255) / 256, 256, 0, stream>>>(ei, deg, h_lin, h_agg);
    k_pool<<<(N_NODES * 32 + 255) / 256, 256, 0, stream>>>(batch, h_agg, gsum, cnt);
    k_mean<<<(initN + 255) / 256, 256, 0, stream>>>(gsum, cnt, gmean);
    k_mlp1<<<64, 128, 0, stream>>>(gmean, wl, bl, g2);        // 256 tiles, 4 waves/blk
    k_head<<<NUM_GRAPHS / 16, 128, 0, stream>>>(g2, w2, b2, out);
}